// GCN_83356725280823
// MI455X (gfx1250) — compile-verified
//
#include <hip/hip_runtime.h>

// GCN 2-layer inference for MI455X (gfx1250).
// Dense GEMMs use V_WMMA_F32_16X16X4_F32 (true fp32 matrix op -> matches the
// fp32 reference numerics). A-tiles are staged into LDS with the CDNA5
// GLOBAL_LOAD_ASYNC_TO_LDS_B128 path (ASYNCcnt-tracked, no VGPR round trip).
// Edge aggregation is the memory-bound part (~2.5 GB of gather + f32 atomics
// -> ~110us at 23.3 TB/s HBM); it runs one wave32 per edge with scalar
// (SMEM) index/degree loads so the vector pipe carries only payload+atomics.

#define IN_F   256
#define H1_F   128
#define H2_F   64
#define BN_EPS 1e-5f

typedef __attribute__((ext_vector_type(2))) float v2f;
typedef __attribute__((ext_vector_type(8))) float v8f;

// ---------------------------------------------------------------- degrees ---
__global__ void init_deg_kernel(float* __restrict__ deg, int n) {
  int i = blockIdx.x * blockDim.x + threadIdx.x;
  if (i < n) deg[i] = 1.0f;                    // +1 self-loop
}

// edge_index declared int64 in the reference -> long long here.
__global__ void accum_deg_kernel(const long long* __restrict__ dst,
                                 float* __restrict__ deg, int e) {
  int i = blockIdx.x * blockDim.x + threadIdx.x;
  if (i < e) atomicAdd(&deg[(int)dst[i]], 1.0f);
}

__global__ void rsqrt_kernel(const float* __restrict__ deg,
                             float* __restrict__ dinv, int n) {
  int i = blockIdx.x * blockDim.x + threadIdx.x;
  if (i < n) dinv[i] = rsqrtf(deg[i]);
}

// ------------------------------------------------------------- WMMA GEMM ----
// C[M x NCOL] = A[M x K] @ B[K x NCOL], all row-major fp32.
// One wave computes one 16x16 C tile via K/4 chained V_WMMA_F32_16X16X4_F32.
// A 16xK tile is staged in LDS via GLOBAL_LOAD_ASYNC_TO_LDS_B128 with +4
// float row padding (K+4 elem stride: bank = (m*4 + k) % 64 -> conflict-free
// ds_load_b64 per lane).
//
// Fragment layouts per CDNA5 ISA 7.12.2:
//   A (16x4, 2 VGPR): lanes 0-15 -> {K=0,K=1} of row M=lane;
//                     lanes 16-31 -> {K=2,K=3} of row M=lane-16.
//   B (4x16, 2 VGPR): mirrored (lanes 0-15 rows K=0,1; lanes 16-31 rows K=2,3).
//   C/D (16x16, 8 VGPR): VGPR j = row j (lanes 0-15) / row j+8 (lanes 16-31).
template <int K, int NCOL>
__global__ __launch_bounds__(32 * (NCOL / 16)) void gemm_wmma_f32(
    const float* __restrict__ A, const float* __restrict__ B,
    float* __restrict__ C, int M) {
  constexpr int LDA = K + 4;
  __shared__ float As[16 * LDA];

  const int lane = threadIdx.x & 31;
  const int wave = threadIdx.x >> 5;
  const int row0 = blockIdx.x * 16;
  const int col0 = wave * 16;
  constexpr int NTHREADS = 32 * (NCOL / 16);

  // Async (memory -> LDS) staging of the 16 x K A tile, 16B per lane-op.
  constexpr int NV4 = 16 * K / 4;
#pragma unroll
  for (int v = threadIdx.x; v < NV4; v += NTHREADS) {
    int r  = v / (K / 4);
    int c4 = (v % (K / 4)) * 4;
    unsigned goff = (unsigned)(((unsigned)(row0 + r) * (unsigned)K + c4) * 4u);
    unsigned laddr = (unsigned)(size_t)(&As[r * LDA + c4]);   // LDS byte addr
    asm volatile("global_load_async_to_lds_b128 %0, %1, %2"
                 :: "v"(laddr), "v"(goff), "s"(A) : "memory");
  }
  asm volatile("s_wait_asynccnt 0x0" ::: "memory");
  __syncthreads();

  const int mhalf = lane & 15;            // row within tile for A frag
  const int koff  = (lane >> 4) * 2;      // K sub-offset: 0 or 2
  const int ncol  = col0 + (lane & 15);   // global column for B/C frag

  v8f acc = {0.f, 0.f, 0.f, 0.f, 0.f, 0.f, 0.f, 0.f};

#pragma unroll 4
  for (int kk = 0; kk < K; kk += 4) {
    v2f a = *(const v2f*)(&As[mhalf * LDA + kk + koff]);   // ds_load_b64
    v2f b;
    b.x = B[(size_t)(kk + koff) * NCOL + ncol];
    b.y = B[(size_t)(kk + koff + 1) * NCOL + ncol];
    acc = __builtin_amdgcn_wmma_f32_16x16x4_f32(
        /*neg_a=*/false, a, /*neg_b=*/false, b,
        /*c_mod=*/(short)0, acc, /*reuse_a=*/false, /*reuse_b=*/false);
  }

  const int mbase = row0 + ((lane >> 4) << 3);   // +0 or +8
#pragma unroll
  for (int j = 0; j < 8; ++j)
    C[(size_t)(mbase + j) * NCOL + ncol] = acc[j];
}

// ------------------------------------------------- self-loop initialization -
// agg[n][f] = xw[n][f] / deg[n]  (= xw * dinv^2), float4 vectorized.
__global__ void selfloop_kernel(const float* __restrict__ xw,
                                const float* __restrict__ dinv,
                                float* __restrict__ agg,
                                int total4, int quadsPerRow) {
  int i = blockIdx.x * blockDim.x + threadIdx.x;
  if (i >= total4) return;
  int node = i / quadsPerRow;
  float d = dinv[node];
  float s = d * d;
  float4 v = *(const float4*)(xw + (size_t)i * 4);
  v.x *= s; v.y *= s; v.z *= s; v.w *= s;
  *(float4*)(agg + (size_t)i * 4) = v;
}

// --------------------------------------------------------- edge scatter-add -
// One wave32 per edge; lane handles CHUNK floats (4 for H=128, 2 for H=64).
// Edge id is forced into an SGPR (readfirstlane) so src/dst/dinv loads become
// scalar loads; the vector pipe carries only the payload + f32 atomics.
template <int CHUNK>
__global__ void scatter_kernel(const float* __restrict__ xw,
                               const long long* __restrict__ src,
                               const long long* __restrict__ dst,
                               const float* __restrict__ dinv,
                               float* __restrict__ agg, int nE, int h) {
  int e = blockIdx.x * (blockDim.x >> 5) + (threadIdx.x >> 5);
  e = __builtin_amdgcn_readfirstlane(e);       // wave-uniform -> SGPR
  if (e >= nE) return;
  const int lane = threadIdx.x & 31;
  const int s = (int)src[e];                   // scalar (SMEM) loads
  const int d = (int)dst[e];
  const float norm = dinv[s] * dinv[d];
  const float* xp = xw + (size_t)s * h + lane * CHUNK;
  float*       ap = agg + (size_t)d * h + lane * CHUNK;
  if constexpr (CHUNK == 4) {
    float4 m = *(const float4*)xp;
    atomicAdd(ap + 0, m.x * norm);
    atomicAdd(ap + 1, m.y * norm);
    atomicAdd(ap + 2, m.z * norm);
    atomicAdd(ap + 3, m.w * norm);
  } else {
    float2 m = *(const float2*)xp;
    atomicAdd(ap + 0, m.x * norm);
    atomicAdd(ap + 1, m.y * norm);
  }
}

// ---------------------------------------------- bias + BatchNorm + ReLU -----
__global__ void bn_relu_kernel(const float* __restrict__ agg,
                               const float* __restrict__ bias,
                               const float* __restrict__ gamma,
                               const float* __restrict__ beta,
                               const float* __restrict__ mean,
                               const float* __restrict__ var,
                               float* __restrict__ out, int total, int hmask) {
  int i = blockIdx.x * blockDim.x + threadIdx.x;
  if (i >= total) return;
  int f = i & hmask;
  float x  = agg[i] + bias[f];
  float xh = (x - mean[f]) * rsqrtf(var[f] + BN_EPS);
  out[i] = fmaxf(xh * gamma[f] + beta[f], 0.0f);
}

// ------------------------- bias + BN + ReLU + final 64x2 linear, fused ------
__global__ void final_kernel(const float* __restrict__ agg2,
                             const float* __restrict__ b2,
                             const float* __restrict__ g2,
                             const float* __restrict__ be2,
                             const float* __restrict__ m2,
                             const float* __restrict__ v2,
                             const float* __restrict__ Wl,
                             const float* __restrict__ bl,
                             float* __restrict__ out, int n) {
  int i = blockIdx.x * blockDim.x + threadIdx.x;
  if (i >= n) return;
  const float* row = agg2 + (size_t)i * H2_F;
  float o0 = bl[0], o1 = bl[1];
#pragma unroll 8
  for (int f = 0; f < H2_F; ++f) {
    float x  = row[f] + b2[f];
    float xh = (x - m2[f]) * rsqrtf(v2[f] + BN_EPS);
    float hr = fmaxf(xh * g2[f] + be2[f], 0.0f);
    o0 = fmaf(hr, Wl[f * 2 + 0], o0);
    o1 = fmaf(hr, Wl[f * 2 + 1], o1);
  }
  out[(size_t)i * 2 + 0] = o0;
  out[(size_t)i * 2 + 1] = o1;
}

// ---------------------------------------------------------------------------
extern "C" void kernel_launch(void* const* d_in, const int* in_sizes, int n_in,
                              void* d_out, int out_size, void* d_ws,
                              size_t ws_size, hipStream_t stream) {
  (void)n_in; (void)out_size; (void)ws_size;

  const float*     x   = (const float*)d_in[0];
  const long long* ei  = (const long long*)d_in[1];   // int64 per reference
  const float*     W1  = (const float*)d_in[2];
  const float*     b1  = (const float*)d_in[3];
  const float*     g1  = (const float*)d_in[4];
  const float*     be1 = (const float*)d_in[5];
  const float*     m1  = (const float*)d_in[6];
  const float*     v1  = (const float*)d_in[7];
  const float*     W2  = (const float*)d_in[8];
  const float*     b2  = (const float*)d_in[9];
  const float*     g2  = (const float*)d_in[10];
  const float*     be2 = (const float*)d_in[11];
  const float*     m2  = (const float*)d_in[12];
  const float*     v2  = (const float*)d_in[13];
  const float*     Wl  = (const float*)d_in[14];
  const float*     bl  = (const float*)d_in[15];
  float*           out = (float*)d_out;

  const int nN = in_sizes[0] / IN_F;          // 50000 (divisible by 16)
  const int nE = in_sizes[1] / 2;             // 1,600,000
  const long long* src = ei;
  const long long* dst = ei + nE;

  // Workspace carve (floats): deg | dinv | bufA(xw1->h1) | bufB(agg1)
  //                           | bufC(xw2) | bufD(agg2)   (~77 MB total)
  float* ws   = (float*)d_ws;
  float* deg  = ws;                          ws += nN;
  float* dinv = ws;                          ws += nN;
  float* bufA = ws;                          ws += (size_t)nN * H1_F;
  float* bufB = ws;                          ws += (size_t)nN * H1_F;
  float* bufC = ws;                          ws += (size_t)nN * H2_F;
  float* bufD = ws;

  const int TB = 256;
  const int wavesPerBlock = TB / 32;
  // 1-3) degrees: deg = in-degree(dst) + 1 ; dinv = deg^-1/2
  init_deg_kernel<<<(nN + TB - 1) / TB, TB, 0, stream>>>(deg, nN);
  accum_deg_kernel<<<(nE + TB - 1) / TB, TB, 0, stream>>>(dst, deg, nE);
  rsqrt_kernel<<<(nN + TB - 1) / TB, TB, 0, stream>>>(deg, dinv, nN);

  // 4) xw1 = x @ W1   (WMMA fp32, 8 waves/block cover N=128)
  gemm_wmma_f32<IN_F, H1_F><<<nN / 16, 256, 0, stream>>>(x, W1, bufA, nN);

  // 5) agg1 = xw1 / deg (self-loop term)
  {
    int total4 = nN * (H1_F / 4);
    selfloop_kernel<<<(total4 + TB - 1) / TB, TB, 0, stream>>>(
        bufA, dinv, bufB, total4, H1_F / 4);
  }
  // 6) agg1 += scatter_add(xw1[src] * dinv[src]*dinv[dst])  (wave per edge)
  scatter_kernel<4><<<(nE + wavesPerBlock - 1) / wavesPerBlock, TB, 0, stream>>>(
      bufA, src, dst, dinv, bufB, nE, H1_F);
  // 7) h1 = relu(bn(agg1 + b1))   (overwrites xw1 buffer — no longer needed)
  {
    int total = nN * H1_F;
    bn_relu_kernel<<<(total + TB - 1) / TB, TB, 0, stream>>>(
        bufB, b1, g1, be1, m1, v1, bufA, total, H1_F - 1);
  }

  // 8) xw2 = h1 @ W2   (WMMA fp32, 4 waves/block cover N=64)
  gemm_wmma_f32<H1_F, H2_F><<<nN / 16, 128, 0, stream>>>(bufA, W2, bufC, nN);

  // 9) agg2 = xw2 / deg
  {
    int total4 = nN * (H2_F / 4);
    selfloop_kernel<<<(total4 + TB - 1) / TB, TB, 0, stream>>>(
        bufC, dinv, bufD, total4, H2_F / 4);
  }
  // 10) agg2 += scatter  (wave per edge, float2 per lane)
  scatter_kernel<2><<<(nE + wavesPerBlock - 1) / wavesPerBlock, TB, 0, stream>>>(
      bufC, src, dst, dinv, bufD, nE, H2_F);
  // 11) out = relu(bn(agg2 + b2)) @ Wl + bl   (fused)
  final_kernel<<<(nN + TB - 1) / TB, TB, 0, stream>>>(
      bufD, b2, g2, be2, m2, v2, Wl, bl, out, nN);
}